// LocalMean_2044404433638
// MI455X (gfx1250) — compile-verified
//
#include <hip/hip_runtime.h>
#include <cstdint>

// 5x5 local mean with reflect padding over 32x3x512x512 fp32.
// Pure bandwidth-bound (~1 FLOP/byte, 201 MB min traffic -> ~8.6us @ 23.3TB/s).
// CDNA5 path: b128 async global->LDS staging (ASYNCcnt), separable sliding
// sums in LDS, non-temporal b128 stores. WMMA intentionally unused: a matmul
// formulation of the stencil multiplies FLOPs without reducing bytes.

#define IMG_W 512
#define IMG_H 512
#define TILE_W 128
#define TILE_H 32
#define IN_H 36      // TILE_H + 4 halo rows
#define RAW_W 136    // staged row: image cols tx0-4 .. tx0+131 (16B-aligned chunks)
#define RAW_S 136    // LDS row stride (floats); row stride = 544B = 34*16B
#define NCHUNK4 34   // 136/4 float4 chunks per row
#define NCHUNK (NCHUNK4 * IN_H)  // 1224 b128 transfers per tile
#define HS_S 132     // LDS row stride for horizontal sums (floats)

typedef float f32x4 __attribute__((ext_vector_type(4)));  // native vector for NT store

__device__ __forceinline__ int reflect512(int i) {
    i = (i < 0) ? -i : i;                                // low edge (edge not repeated)
    return (i > IMG_H - 1) ? (2 * (IMG_H - 1) - i) : i;  // high edge
}

// CDNA5 async global->LDS copies (tracked by ASYNCcnt). lds_off = byte offset
// within the workgroup LDS allocation (low 32 bits of a __shared__ flat addr).
__device__ __forceinline__ void async_ld_b128(uint32_t lds_off, const float* g) {
    asm volatile("global_load_async_to_lds_b128 %0, %1, off"
                 :: "v"(lds_off), "v"(g) : "memory");
}
__device__ __forceinline__ void async_ld_b32(uint32_t lds_off, const float* g) {
    asm volatile("global_load_async_to_lds_b32 %0, %1, off"
                 :: "v"(lds_off), "v"(g) : "memory");
}

__global__ __launch_bounds__(256) void local_mean5_kernel(
        const float* __restrict__ src, float* __restrict__ dst) {
    __shared__ float raw[IN_H * RAW_S];  // staged input tile (+halo), col0 = tx0-4
    __shared__ float hs[IN_H * HS_S];    // horizontal 5-sums, col j = image col tx0+j

    const int tid = threadIdx.x;
    const int tx0 = blockIdx.x * TILE_W;
    const int ty0 = blockIdx.y * TILE_H;
    const size_t plane = (size_t)blockIdx.z * (size_t)(IMG_W * IMG_H);
    const float* sp = src + plane;

    const uint32_t raw_base = (uint32_t)(uintptr_t)&raw[0];

    // ---- Stage tile + halo into LDS: b128 async bulk, scalar reflect at x-edges ----
    for (int e = tid; e < NCHUNK; e += 256) {
        const int r = e / NCHUNK4;
        const int c4 = e - r * NCHUNK4;
        const int gr = reflect512(ty0 - 2 + r);            // row reflect: whole row stays contiguous
        const int col0 = tx0 - 4 + 4 * c4;                 // image col of chunk start
        const uint32_t lds_off = raw_base + 4u * (uint32_t)(r * RAW_S + 4 * c4);
        const float* grow = sp + (size_t)gr * IMG_W;
        if (col0 >= 0 && col0 <= IMG_W - 4) {
            async_ld_b128(lds_off, grow + col0);           // 16B/lane, 512B/wave
        } else {                                           // <=1 edge chunk per row
            #pragma unroll
            for (int i = 0; i < 4; ++i)
                async_ld_b32(lds_off + 4u * i, grow + reflect512(col0 + i));
        }
    }
    asm volatile("s_wait_asynccnt 0" ::: "memory");
    __syncthreads();

    // ---- Pass 1: horizontal 5-sums (sliding window, 4 cols/thread/row) ----
    {
        const int c = (tid & 31) * 4;                      // hs col base, 0..124
        for (int r = (tid >> 5); r < IN_H; r += 8) {
            // hs[c+k] = sum raw cols (c+k+2)..(c+k+6)  (raw col j = image col tx0-4+j)
            const float* rp = &raw[r * RAW_S + c + 2];
            float w0 = rp[0], w1 = rp[1], w2 = rp[2], w3 = rp[3];
            float w4 = rp[4], w5 = rp[5], w6 = rp[6], w7 = rp[7];
            float s0 = w0 + w1 + w2 + w3 + w4;
            float s1 = s0 - w0 + w5;
            float s2 = s1 - w1 + w6;
            float s3 = s2 - w2 + w7;
            float* hp = &hs[r * HS_S + c];
            hp[0] = s0; hp[1] = s1; hp[2] = s2; hp[3] = s3;
        }
    }
    __syncthreads();

    // ---- Pass 2: vertical 5-sums + scale, 4x4 outputs/thread ----
    const float inv25 = 1.0f / 25.0f;
    const int cx = (tid & 31) * 4;  // out col base within tile, 0..124
    const int ry = (tid >> 5) * 4;  // out row base within tile, 0..28
    float o[4][4];
    #pragma unroll
    for (int i = 0; i < 4; ++i) {
        const float* cp = &hs[ry * HS_S + cx + i];  // out row y needs hs rows y..y+4
        float h0 = cp[0 * HS_S], h1 = cp[1 * HS_S], h2 = cp[2 * HS_S], h3 = cp[3 * HS_S];
        float h4 = cp[4 * HS_S], h5 = cp[5 * HS_S], h6 = cp[6 * HS_S], h7 = cp[7 * HS_S];
        float v = h0 + h1 + h2 + h3 + h4;
        o[0][i] = v * inv25;
        v += h5 - h0; o[1][i] = v * inv25;
        v += h6 - h1; o[2][i] = v * inv25;
        v += h7 - h2; o[3][i] = v * inv25;
    }
    float* dp = dst + plane + (size_t)(ty0 + ry) * IMG_W + (tx0 + cx);
    #pragma unroll
    for (int r = 0; r < 4; ++r) {
        f32x4 val = { o[r][0], o[r][1], o[r][2], o[r][3] };
        // Output is never re-read: keep it out of L2 so input halo rows stay resident.
        __builtin_nontemporal_store(val, (f32x4*)(dp + (size_t)r * IMG_W));
    }
}

extern "C" void kernel_launch(void* const* d_in, const int* in_sizes, int n_in,
                              void* d_out, int out_size, void* d_ws, size_t ws_size,
                              hipStream_t stream) {
    const float* src = (const float*)d_in[0];
    float* dst = (float*)d_out;
    const int planes = in_sizes[0] / (IMG_W * IMG_H);  // 32*3 = 96
    dim3 grid(IMG_W / TILE_W, IMG_H / TILE_H, planes);
    dim3 block(256);
    hipLaunchKernelGGL(local_mean5_kernel, grid, block, 0, stream, src, dst);
}